// OnlineTripletLoss_16475494547623
// MI455X (gfx1250) — compile-verified
//
#include <hip/hip_runtime.h>
#include <hip/hip_bf16.h>

#define B_N   4096
#define D_DIM 128
#define NWAVES 8
#define EPS    1e-6f
#define NEG_INF_F (-1e30f)
#define POS_INF_F (1e30f)

typedef __attribute__((ext_vector_type(2))) float v2f;
typedef __attribute__((ext_vector_type(8))) float v8f;

// ---------------- Kernel 1: per-row sum of squares and sum -------------------
__global__ __launch_bounds__(256) void k_rowstats(const float* __restrict__ E,
                                                  float* __restrict__ sq,
                                                  float* __restrict__ s) {
  int row  = blockIdx.x * 8 + (threadIdx.x >> 5);
  int lane = threadIdx.x & 31;
  const float4 x = *(const float4*)(E + (size_t)row * D_DIM + lane * 4);
  float ss = x.x * x.x + x.y * x.y + x.z * x.z + x.w * x.w;
  float sm = x.x + x.y + x.z + x.w;
  #pragma unroll
  for (int m = 16; m >= 1; m >>= 1) {
    ss += __shfl_xor(ss, m, 32);
    sm += __shfl_xor(sm, m, 32);
  }
  if (lane == 0) { sq[row] = ss; s[row] = sm; }
}

// ---- Kernel 2: fused fp32-WMMA Gram tiles + masked argmax/argmin ------------
// One block = 16 rows.  8 waves; wave w handles N-tiles w, w+8, ...
__global__ __launch_bounds__(256) void k_mine(const float* __restrict__ E,
                                              const int* __restrict__ tgt,
                                              const unsigned char* __restrict__ posM,
                                              const unsigned char* __restrict__ negM,
                                              const float* __restrict__ sq,
                                              const float* __restrict__ s,
                                              int* __restrict__ posIdx,
                                              int* __restrict__ negIdx,
                                              int* __restrict__ validOut) {
  const int tid  = threadIdx.x;
  const int wave = tid >> 5;
  const int lane = tid & 31;
  const int hi   = lane >> 4;   // 0: K-pair {0,1} / rows 0..7 ; 1: K-pair {2,3} / rows 8..15
  const int ln   = lane & 15;
  const int rowBase = blockIdx.x * 16;

  // A fragments for all K (16x4 tiles): lane = M, khalf by lane-half.
  const float* arow = E + (size_t)(rowBase + ln) * D_DIM;
  v2f a[32];
  #pragma unroll
  for (int kk = 0; kk < 32; ++kk)
    a[kk] = *(const v2f*)(arow + kk * 4 + hi * 2);

  float sqi[8], si[8];
  #pragma unroll
  for (int v = 0; v < 8; ++v) {
    int m = v + 8 * hi;
    sqi[v] = sq[rowBase + m];
    si[v]  = s[rowBase + m];
  }

  float bestP[8], bestN[8];
  int   bpi[8],  bni[8];
  #pragma unroll
  for (int v = 0; v < 8; ++v) {
    bestP[v] = NEG_INF_F; bpi[v] = 0;
    bestN[v] = POS_INF_F; bni[v] = 0;
  }

  const float epsq = (float)D_DIM * (EPS * EPS);

  for (int t = wave; t < B_N / 16; t += NWAVES) {
    const int j = t * 16 + ln;                    // this lane's column
    const float* brow = E + (size_t)j * D_DIM;    // B = e^T: B[k][n] = e[j][k]
    v8f c = {};
    #pragma unroll
    for (int kk = 0; kk < 32; ++kk) {
      v2f b = *(const v2f*)(brow + kk * 4 + hi * 2);
      c = __builtin_amdgcn_wmma_f32_16x16x4_f32(false, a[kk], false, b,
                                                (short)0, c, false, false);
    }
    const int   tj  = tgt[j];
    const float sqj = sq[j];
    const float sj  = s[j];
    #pragma unroll
    for (int v = 0; v < 8; ++v) {
      const int i  = rowBase + v + 8 * hi;
      float d2 = sqi[v] + sqj - 2.0f * c[v] + 2.0f * EPS * (si[v] - sj) + epsq;
      float dm = __builtin_sqrtf(fmaxf(d2, 0.0f));
      const size_t mi = (size_t)i * B_N + (size_t)tj;
      if (posM[mi] && dm > bestP[v]) { bestP[v] = dm; bpi[v] = j; }
      if (negM[mi] && dm < bestN[v]) { bestN[v] = dm; bni[v] = j; }
    }
  }

  // Reduce across the 16 lanes of each half (tie-break: smaller index first,
  // matching JAX argmax/argmin first-occurrence semantics).
  #pragma unroll
  for (int v = 0; v < 8; ++v) {
    #pragma unroll
    for (int m = 1; m < 16; m <<= 1) {
      float ov = __shfl_xor(bestP[v], m, 32);
      int   oi = __shfl_xor(bpi[v],  m, 32);
      if (ov > bestP[v] || (ov == bestP[v] && oi < bpi[v])) { bestP[v] = ov; bpi[v] = oi; }
      ov = __shfl_xor(bestN[v], m, 32);
      oi = __shfl_xor(bni[v],  m, 32);
      if (ov < bestN[v] || (ov == bestN[v] && oi < bni[v])) { bestN[v] = ov; bni[v] = oi; }
    }
  }

  __shared__ float sPv[NWAVES][16], sNv[NWAVES][16];
  __shared__ int   sPi[NWAVES][16], sNi[NWAVES][16];
  if (ln == 0) {
    #pragma unroll
    for (int v = 0; v < 8; ++v) {
      int m = v + 8 * hi;
      sPv[wave][m] = bestP[v]; sPi[wave][m] = bpi[v];
      sNv[wave][m] = bestN[v]; sNi[wave][m] = bni[v];
    }
  }
  __syncthreads();

  if (tid < 16) {
    float pv = NEG_INF_F, nv = POS_INF_F;
    int   pi = 0, ni = 0;
    for (int w = 0; w < NWAVES; ++w) {
      float cv = sPv[w][tid]; int ci = sPi[w][tid];
      if (cv > pv || (cv == pv && ci < pi)) { pv = cv; pi = ci; }
      cv = sNv[w][tid]; ci = sNi[w][tid];
      if (cv < nv || (cv == nv && ci < ni)) { nv = cv; ni = ci; }
    }
    const int i = rowBase + tid;
    posIdx[i]   = pi;
    negIdx[i]   = ni;
    validOut[i] = (pv > -1e29f && nv < 1e29f) ? 1 : 0;
  }
}

// ---------------- Kernel 3: per-row triplet loss (deterministic) -------------
__global__ __launch_bounds__(256) void k_triplet(const float* __restrict__ E,
                                                 const int* __restrict__ posIdx,
                                                 const int* __restrict__ negIdx,
                                                 const int* __restrict__ valid,
                                                 float* __restrict__ lossArr,
                                                 float* __restrict__ wArr) {
  int row  = blockIdx.x * 8 + (threadIdx.x >> 5);
  int lane = threadIdx.x & 31;
  const float4 a4 = *(const float4*)(E + (size_t)row * D_DIM + lane * 4);
  const float4 p4 = *(const float4*)(E + (size_t)posIdx[row] * D_DIM + lane * 4);
  const float4 n4 = *(const float4*)(E + (size_t)negIdx[row] * D_DIM + lane * 4);

  float dap, dan, dpn, sap = 0.f, san = 0.f, spn = 0.f;
  dap = a4.x - p4.x + EPS; sap += dap * dap;
  dap = a4.y - p4.y + EPS; sap += dap * dap;
  dap = a4.z - p4.z + EPS; sap += dap * dap;
  dap = a4.w - p4.w + EPS; sap += dap * dap;
  dan = a4.x - n4.x + EPS; san += dan * dan;
  dan = a4.y - n4.y + EPS; san += dan * dan;
  dan = a4.z - n4.z + EPS; san += dan * dan;
  dan = a4.w - n4.w + EPS; san += dan * dan;
  dpn = p4.x - n4.x + EPS; spn += dpn * dpn;
  dpn = p4.y - n4.y + EPS; spn += dpn * dpn;
  dpn = p4.z - n4.z + EPS; spn += dpn * dpn;
  dpn = p4.w - n4.w + EPS; spn += dpn * dpn;

  #pragma unroll
  for (int m = 16; m >= 1; m >>= 1) {
    sap += __shfl_xor(sap, m, 32);
    san += __shfl_xor(san, m, 32);
    spn += __shfl_xor(spn, m, 32);
  }
  if (lane == 0) {
    float ap = __builtin_sqrtf(sap);
    float an = __builtin_sqrtf(san);
    float pn = __builtin_sqrtf(spn);
    float loss = fmaxf(ap - fminf(an, pn) + 1.0f, 0.0f);
    float w = valid[row] ? 1.0f : 0.0f;
    lossArr[row] = loss * w;
    wArr[row]    = w;
  }
}

// ---------------- Kernel 4: fixed-order final reduction ----------------------
__global__ __launch_bounds__(256) void k_finalize(const float* __restrict__ lossArr,
                                                  const float* __restrict__ wArr,
                                                  float* __restrict__ out) {
  __shared__ float sl[256], sw[256];
  float l = 0.f, w = 0.f;
  for (int i = threadIdx.x; i < B_N; i += 256) { l += lossArr[i]; w += wArr[i]; }
  sl[threadIdx.x] = l; sw[threadIdx.x] = w;
  __syncthreads();
  for (int st = 128; st > 0; st >>= 1) {
    if ((int)threadIdx.x < st) {
      sl[threadIdx.x] += sl[threadIdx.x + st];
      sw[threadIdx.x] += sw[threadIdx.x + st];
    }
    __syncthreads();
  }
  if (threadIdx.x == 0) out[0] = sl[0] / fmaxf(sw[0], 1.0f);
}

extern "C" void kernel_launch(void* const* d_in, const int* in_sizes, int n_in,
                              void* d_out, int out_size, void* d_ws, size_t ws_size,
                              hipStream_t stream) {
  const float*         E    = (const float*)d_in[0];
  const int*           tgt  = (const int*)d_in[1];
  const unsigned char* posM = (const unsigned char*)d_in[2];  // jnp.bool_ = 1 byte
  const unsigned char* negM = (const unsigned char*)d_in[3];
  float* out = (float*)d_out;

  float* sq      = (float*)d_ws;                    // 4096
  float* s       = sq + B_N;                        // 4096
  float* lossArr = s + B_N;                         // 4096
  float* wArr    = lossArr + B_N;                   // 4096
  int*   posIdx  = (int*)(wArr + B_N);              // 4096
  int*   negIdx  = posIdx + B_N;                    // 4096
  int*   valid   = negIdx + B_N;                    // 4096

  k_rowstats<<<B_N / 8, 256, 0, stream>>>(E, sq, s);
  k_mine<<<B_N / 16, 256, 0, stream>>>(E, tgt, posM, negM, sq, s,
                                       posIdx, negIdx, valid);
  k_triplet<<<B_N / 8, 256, 0, stream>>>(E, posIdx, negIdx, valid, lossArr, wArr);
  k_finalize<<<1, 256, 0, stream>>>(lossArr, wArr, out);
}